// VitsKLProcessor_28458453303303
// MI455X (gfx1250) — compile-verified
//
#include <hip/hip_runtime.h>
#include <hip/hip_bf16.h>
#include <stdint.h>

// ---------------------------------------------------------------------------
// VITS KL loss + monotonic alignment search for MI455X (gfx1250, wave32).
//   B=16, C=192, Tx=512, Ty=2048
// Pipeline:
//   1. vits_prep  : t_x_len/t_y_len from masks, per-x bias, zero accumulator
//   2. vits_gemm  : log_p[b][y][x] via V_WMMA_F32_16X16X4_F32 (exact fp32),
//                   written into the *path* region of d_out (same size).
//                   LDS tiles are K-pair interleaved so each WMMA fragment is
//                   one aligned ds_load_b64 straight into an even VGPR pair.
//   3. vits_fwd   : serial-over-y Viterbi forward; packs backtrack decision
//                   bits with ballot_w32 (1 bit per (y,x))
//   4. vits_backtrack : per-batch serial backtrack -> xs_of_y
//   5. vits_path  : overwrite log_p region with hard 0/1 path
//   6. vits_zcopy : pass-through z_flow
//   7. vits_loss + vits_finalize : masked KL reduction -> loss scalar
// ---------------------------------------------------------------------------

typedef __attribute__((ext_vector_type(2))) float v2f;
typedef __attribute__((ext_vector_type(8))) float v8f;

#define NEGF (-1e9f)

constexpr int B_  = 16;
constexpr int C_  = 192;
constexpr int TX_ = 512;
constexpr int TY_ = 2048;

constexpr int TILE_Y = 64;    // workgroup tile along y (WMMA M)
constexpr int TILE_X = 128;   // workgroup tile along x (WMMA N)
constexpr int KC     = 16;    // K-chunk staged in LDS (4 WMMA k-steps)
// Pad k-pair rows: row stride == 16 DWORDs (mod 64 banks), so the two lane
// halves of a wave (which read different k-pairs) share 16 of the banks.
constexpr int PADY   = 8;     // float2 slots of padding per k-pair row (A side)
constexpr int PADX   = 8;     // float2 slots of padding per k-pair row (B side)

// ---------------------------------------------------------------------------
// 1. prep: sequence lengths, per-x bias = -0.5*(term_c + term_d), zero accum
// ---------------------------------------------------------------------------
__global__ __launch_bounds__(256) void vits_prep(
    const float* __restrict__ ymask, const float* __restrict__ xmask,
    const float* __restrict__ mp, const float* __restrict__ lsp,
    int* __restrict__ txl, int* __restrict__ tyl,
    float* __restrict__ bias, float* __restrict__ accum) {
  const int b = blockIdx.x, tid = threadIdx.x;
  __shared__ float ry[256], rx[256];
  float sy = 0.f, sx = 0.f;
  for (int y = tid; y < TY_; y += 256) sy += ymask[b * TY_ + y];
  for (int x = tid; x < TX_; x += 256) sx += xmask[b * TX_ + x];
  ry[tid] = sy; rx[tid] = sx;
  __syncthreads();
  for (int s = 128; s > 0; s >>= 1) {
    if (tid < s) { ry[tid] += ry[tid + s]; rx[tid] += rx[tid + s]; }
    __syncthreads();
  }
  if (tid == 0) {
    tyl[b] = (int)(ry[0] + 0.5f);
    txl[b] = (int)(rx[0] + 0.5f);
    if (b == 0) *accum = 0.f;
  }
  for (int x = tid; x < TX_; x += 256) {
    float tc = 0.f, td = 0.f;
    for (int c = 0; c < C_; ++c) {
      const int gi = (b * C_ + c) * TX_ + x;
      const float ls = lsp[gi];
      const float mv = mp[gi];
      const float o  = __expf(-2.f * ls);
      tc += mv * mv * o;
      td += 2.f * ls;
    }
    bias[b * TX_ + x] = -0.5f * (tc + td);
  }
}

// ---------------------------------------------------------------------------
// 2. log_p GEMM with fp32 WMMA.
//    log_p[b][y][x] = Sum_c z2[c,y]*(-0.5*o_p[c,x]) + z[c,y]*(m_p*o_p)[c,x]
//                     + bias[b][x]
//    A (M=y, K=c) = z / z^2 fragments; B (K=c, N=x) = transformed text stats.
//    Each wave: 16(y) x 64(x) tile -> 4 v8f accumulators, 384 WMMA per wave.
//    Per the ISA fragment layout, lanes 0-15 carry K={k4,k4+1} and lanes
//    16-31 carry K={k4+2,k4+3}; with K-pair interleaved LDS each fragment is
//    a single contiguous 8-byte read.
// ---------------------------------------------------------------------------
__global__ __launch_bounds__(256) void vits_gemm(
    const float* __restrict__ z, const float* __restrict__ mp,
    const float* __restrict__ lsp, const float* __restrict__ bias,
    float* __restrict__ logp) {
  const int b  = blockIdx.z;
  const int y0 = blockIdx.y * TILE_Y;
  const int x0 = blockIdx.x * TILE_X;

  // [k-pair][row][2 consecutive k values]
  __shared__ float sZ [KC / 2][TILE_Y + PADY][2];   // z
  __shared__ float sZ2[KC / 2][TILE_Y + PADY][2];   // z^2
  __shared__ float sA1[KC / 2][TILE_X + PADX][2];   // -0.5 * o_p
  __shared__ float sA2[KC / 2][TILE_X + PADX][2];   //  m_p * o_p

  const int tid  = threadIdx.x;
  const int w    = tid >> 5;          // wave id 0..7
  const int lane = tid & 31;
  const int wy   = (w >> 1) * 16;     // wave M offset (y)
  const int wx   = (w & 1) * 64;      // wave N offset (x)
  const int hi   = lane >> 4;         // lane half -> which k-pair of the k4 group
  const int l4   = lane & 15;

  v8f acc[4] = {};                    // 4 x (16x16) fp32 accumulators

  for (int kk = 0; kk < C_; kk += KC) {
    // stage audio-side (z, z^2): KC*TILE_Y = 1024 elems
    for (int i = tid; i < KC * TILE_Y; i += 256) {
      const int k = i / TILE_Y, m = i % TILE_Y;
      const float zz = z[(b * C_ + kk + k) * TY_ + y0 + m];
      sZ [k >> 1][m][k & 1] = zz;
      sZ2[k >> 1][m][k & 1] = zz * zz;
    }
    // stage text-side transforms: KC*TILE_X = 2048 elems
    for (int i = tid; i < KC * TILE_X; i += 256) {
      const int k = i / TILE_X, n = i % TILE_X;
      const int gi = (b * C_ + kk + k) * TX_ + x0 + n;
      const float mv = mp[gi];
      const float ls = lsp[gi];
      const float o  = __expf(-2.f * ls);
      sA1[k >> 1][n][k & 1] = -0.5f * o;
      sA2[k >> 1][n][k & 1] = mv * o;
    }
    __syncthreads();

#pragma unroll
    for (int k4 = 0; k4 < KC; k4 += 4) {
      const int kp = (k4 >> 1) + hi;  // k-pair index for this lane half
      // A fragments: one aligned 8B LDS read each
      const v2f az2 = *(const v2f*)(&sZ2[kp][wy + l4][0]);
      const v2f az  = *(const v2f*)(&sZ [kp][wy + l4][0]);
#pragma unroll
      for (int j = 0; j < 4; ++j) {
        const int n = wx + j * 16 + l4;
        const v2f b1 = *(const v2f*)(&sA1[kp][n][0]);
        const v2f b2 = *(const v2f*)(&sA2[kp][n][0]);
        acc[j] = __builtin_amdgcn_wmma_f32_16x16x4_f32(
            false, az2, false, b1, (short)0, acc[j], false, false);
        acc[j] = __builtin_amdgcn_wmma_f32_16x16x4_f32(
            false, az,  false, b2, (short)0, acc[j], false, false);
      }
    }
    __syncthreads();
  }

  // epilogue: C/D layout -> VGPR i: lanes 0-15 M=i, lanes 16-31 M=8+i; N=lane&15
#pragma unroll
  for (int j = 0; j < 4; ++j) {
    const int x = x0 + wx + j * 16 + l4;
    const float bv = bias[b * TX_ + x];
#pragma unroll
    for (int i = 0; i < 8; ++i) {
      const int y = y0 + wy + hi * 8 + i;
      logp[((size_t)(b * TY_ + y)) * TX_ + x] = acc[j][i] + bv;
    }
  }
}

// ---------------------------------------------------------------------------
// 3. forward Viterbi DP: one workgroup per batch, thread == x column.
//    Packs backtrack decision bits (value[y-1][x] < value[y-1][x-1]) via
//    wave32 ballot so the 67 MB value matrix never hits memory.
// ---------------------------------------------------------------------------
__global__ __launch_bounds__(512) void vits_fwd(
    const float* __restrict__ logp, const int* __restrict__ txl,
    const int* __restrict__ tyl, uint32_t* __restrict__ bits) {
  const int b = blockIdx.x;
  const int x = threadIdx.x;
  __shared__ float buf[2][TX_];
  const int tx = txl[b], ty = tyl[b];

  buf[0][x] = NEGF;
  __syncthreads();

  for (int y = 0; y < TY_; ++y) {
    const float* pv = buf[y & 1];
    const float pvx   = pv[x];
    const float pvxm1 = (x > 0) ? pv[x - 1] : NEGF;

    const bool pred = (x > 0) && (pvx < pvxm1);
    const unsigned mask = __builtin_amdgcn_ballot_w32(pred);
    if ((x & 31) == 0)
      bits[((size_t)(b * TY_ + y)) * (TX_ / 32) + (x >> 5)] = mask;

    const float vm1 = (x == 0) ? ((y == 0) ? 0.f : NEGF) : pvxm1;
    const float vc  = (x == y) ? NEGF : pvx;
    const bool valid = (x <= y) && (x < tx) && (y < ty) && (x >= tx - ty + y);

    const float lp = logp[((size_t)(b * TY_ + y)) * TX_ + x];
    if (y + 1 < TY_)
      __builtin_prefetch(&logp[((size_t)(b * TY_ + y + 1)) * TX_ + x], 0, 1);

    buf[(y + 1) & 1][x] = valid ? (lp + fmaxf(vm1, vc)) : NEGF;
    __syncthreads();
  }
}

// ---------------------------------------------------------------------------
// 4. backtrack: one lane per batch element.
// ---------------------------------------------------------------------------
__global__ void vits_backtrack(const uint32_t* __restrict__ bits,
                               const int* __restrict__ txl,
                               const int* __restrict__ tyl,
                               int* __restrict__ xs) {
  const int b = threadIdx.x;
  if (b >= B_) return;
  const int tx = txl[b], ty = tyl[b];
  int index = tx - 1;
  for (int y = TY_ - 1; y >= 0; --y) {
    const bool active = y < ty;
    xs[b * TY_ + y] = active ? index : -1;
    if (active && index != 0) {
      const uint32_t wbits =
          bits[((size_t)(b * TY_ + y)) * (TX_ / 32) + (index >> 5)];
      const bool diag = (wbits >> (index & 31)) & 1u;
      if (index == y || diag) --index;
    }
  }
}

// ---------------------------------------------------------------------------
// 5. scatter hard path [B,Tx,Ty] (overwrites the log_p scratch region)
// ---------------------------------------------------------------------------
__global__ __launch_bounds__(256) void vits_path(const int* __restrict__ xs,
                                                 float* __restrict__ path) {
  const size_t idx = (size_t)blockIdx.x * blockDim.x + threadIdx.x;
  if (idx >= (size_t)B_ * TX_ * TY_) return;
  const int y = (int)(idx % TY_);
  const int x = (int)((idx / TY_) % TX_);
  const int b = (int)(idx / ((size_t)TX_ * TY_));
  path[idx] = (xs[b * TY_ + y] == x) ? 1.0f : 0.0f;
}

// ---------------------------------------------------------------------------
// 6. z_flow pass-through
// ---------------------------------------------------------------------------
__global__ __launch_bounds__(256) void vits_zcopy(const float* __restrict__ z,
                                                  float* __restrict__ out) {
  const size_t i = (size_t)blockIdx.x * blockDim.x + threadIdx.x;
  if (i < (size_t)B_ * C_ * TY_) out[i] = z[i];
}

// ---------------------------------------------------------------------------
// 7. masked KL reduction + finalize
// ---------------------------------------------------------------------------
__global__ __launch_bounds__(256) void vits_loss(
    const float* __restrict__ z, const float* __restrict__ mp,
    const float* __restrict__ lsp, const int* __restrict__ xs,
    const int* __restrict__ tyl, float* __restrict__ accum) {
  __shared__ float red[256];
  const int t = blockIdx.x * blockDim.x + threadIdx.x;
  float local = 0.f;
  if (t < B_ * TY_) {
    const int b = t / TY_, y = t % TY_;
    if (y < tyl[b]) {
      const int x = xs[t];
      for (int c = 0; c < C_; ++c) {
        const float ls = lsp[(b * C_ + c) * TX_ + x];
        const float mv = mp [(b * C_ + c) * TX_ + x];
        const float zv = z  [(b * C_ + c) * TY_ + y];
        const float d  = zv - mv;
        local += ls + 0.5f * __expf(-2.f * ls) * d * d;
      }
    }
  }
  red[threadIdx.x] = local;
  __syncthreads();
  for (int s = 128; s > 0; s >>= 1) {
    if (threadIdx.x < s) red[threadIdx.x] += red[threadIdx.x + s];
    __syncthreads();
  }
  if (threadIdx.x == 0) atomicAdd(accum, red[0]);
}

__global__ void vits_finalize(const float* __restrict__ accum,
                              const float* __restrict__ logdet,
                              const int* __restrict__ tyl,
                              float* __restrict__ out) {
  float ld = 0.f;
  int denom = 0;
  for (int b = 0; b < B_; ++b) { ld += logdet[b]; denom += tyl[b]; }
  out[0] = (*accum - ld) / (float)denom;
}

// ---------------------------------------------------------------------------
// launch
// ---------------------------------------------------------------------------
extern "C" void kernel_launch(void* const* d_in, const int* in_sizes, int n_in,
                              void* d_out, int out_size, void* d_ws,
                              size_t ws_size, hipStream_t stream) {
  const float* z      = (const float*)d_in[0];  // [B,C,Ty]
  const float* mp     = (const float*)d_in[1];  // [B,C,Tx]
  const float* lsp    = (const float*)d_in[2];  // [B,C,Tx]
  const float* logdet = (const float*)d_in[3];  // [B]
  const float* ymask  = (const float*)d_in[4];  // [B,1,Ty]
  const float* xmask  = (const float*)d_in[5];  // [B,1,Tx]

  float* out  = (float*)d_out;
  float* logp = out + 1;                               // path region doubles as log_p scratch
  float* path = out + 1;
  float* outz = out + 1 + (size_t)B_ * TX_ * TY_;

  char* ws = (char*)d_ws;
  uint32_t* bits = (uint32_t*)(ws);                    // 2,097,152 B
  int*      xs   = (int*)     (ws + 2097152);          //   131,072 B
  float*    bias = (float*)   (ws + 2097152 + 131072); //    32,768 B
  int*      txl  = (int*)     (ws + 2260992);
  int*      tyl  = (int*)     (ws + 2260992 + 64);
  float*    accum= (float*)   (ws + 2260992 + 128);

  vits_prep<<<dim3(B_), dim3(256), 0, stream>>>(ymask, xmask, mp, lsp,
                                                txl, tyl, bias, accum);
  vits_gemm<<<dim3(TX_ / TILE_X, TY_ / TILE_Y, B_), dim3(256), 0, stream>>>(
      z, mp, lsp, bias, logp);
  vits_fwd<<<dim3(B_), dim3(TX_), 0, stream>>>(logp, txl, tyl, bits);
  vits_backtrack<<<1, B_, 0, stream>>>(bits, txl, tyl, xs);
  vits_path<<<(int)(((size_t)B_ * TX_ * TY_ + 255) / 256), 256, 0, stream>>>(
      xs, path);
  vits_zcopy<<<(int)(((size_t)B_ * C_ * TY_ + 255) / 256), 256, 0, stream>>>(
      z, outz);
  vits_loss<<<(B_ * TY_ + 255) / 256, 256, 0, stream>>>(z, mp, lsp, xs, tyl,
                                                        accum);
  vits_finalize<<<1, 1, 0, stream>>>(accum, logdet, tyl, out);
}